// RF_ANFISModel_11622181503045
// MI455X (gfx1250) — compile-verified
//
#include <hip/hip_runtime.h>

typedef __attribute__((ext_vector_type(2))) float v2f;
typedef __attribute__((ext_vector_type(8))) float v8f;

#define BATCH   4096
#define N_IN    6
#define N_MF    4
#define N_RULES 4096
#define WAVES   8
#define TILE_M  16

// out[b] = (sum_i S[b][i]*xplus[b][i]) / max(S[b][7], 1e-12)
// S = Rules(B x R) @ CoeffAug(R x 16) via V_WMMA_F32_16X16X4_F32.
// Rules A-fragments generated on the fly from factored membership products;
// CoeffAug B-fragments staged per-256-K-slice into LDS in WMMA lane layout.
__launch_bounds__(256, 1)
__global__ void anfis_wmma_kernel(const float* __restrict__ x,
                                  const float* __restrict__ mfa,
                                  const float* __restrict__ mfb,
                                  const float* __restrict__ mfc,
                                  const float* __restrict__ coeff,
                                  float* __restrict__ out)
{
    __shared__ float  A01[WAVES][TILE_M][16];  // memb0 x memb1 products per row
    __shared__ float  A23[WAVES][TILE_M][16];  // memb2 x memb3 products per row
    __shared__ float2 Bfrag[64][32];           // one 256-K slice of B fragments
    __shared__ float  Sst[WAVES][TILE_M][16];  // C-tile staging for epilogue

    const int tid  = threadIdx.x;
    const int wave = tid >> 5;
    const int lane = tid & 31;
    const int m    = lane & 15;   // batch row within tile
    const int half = lane >> 4;   // K-parity half per WMMA layout
    const int rowbase = blockIdx.x * (WAVES * TILE_M) + wave * TILE_M;
    const int row = rowbase + m;

    // ---- load this row's inputs (lane L and L+16 duplicate row m) ----
    float xv[N_IN];
#pragma unroll
    for (int i = 0; i < N_IN; ++i) xv[i] = x[row * N_IN + i];

    // ---- generalized-bell memberships: 1 / (1 + (((x-c)/a)^2)^b) ----
    float memb[N_IN][N_MF];
#pragma unroll
    for (int v = 0; v < N_IN; ++v) {
#pragma unroll
        for (int j = 0; j < N_MF; ++j) {
            float d    = (xv[v] - mfc[v * N_MF + j]) / mfa[v * N_MF + j];
            float dist = d * d;
            float p    = __powf(dist, mfb[v * N_MF + j]);
            memb[v][j] = 1.0f / (1.0f + p);
        }
    }

    // ---- factored rule tables: rules[r] = A01[r>>8] * A23[(r>>4)&15] * A45[r&15]
    if (half == 0) {
#pragma unroll
        for (int ja = 0; ja < 4; ++ja) {
#pragma unroll
            for (int jb = 0; jb < 4; ++jb) {
                A01[wave][m][ja * 4 + jb] = memb[0][ja] * memb[1][jb];
                A23[wave][m][ja * 4 + jb] = memb[2][ja] * memb[3][jb];
            }
        }
    }
    // A45 factor kept in registers; lane-half picks K-parity (j5 = 2*half+{0,1})
    const float m5a = half ? memb[5][2] : memb[5][0];
    const float m5b = half ? memb[5][3] : memb[5][1];
    float e0[4], e1[4];
#pragma unroll
    for (int s = 0; s < 4; ++s) {
        e0[s] = memb[4][s] * m5a;   // A45[s*4 + 2*half + 0]
        e1[s] = memb[4][s] * m5b;   // A45[s*4 + 2*half + 1]
    }

    // ---- per-thread constants for B-slice staging (n fixed per thread) ----
    const int   sn    = tid & 15;                    // staged column
    const bool  slt7  = (sn < 7);
    const float scaug = (sn == 7) ? 1.0f : 0.0f;     // ones column / zero pad
    const int   skap  = ((tid >> 4) & 1) << 1;       // K-parity of staged pair

    v8f acc = {};

    for (int j01 = 0; j01 < 16; ++j01) {
        // ---- cooperative staging of this slice's B fragments into LDS ----
        __syncthreads();   // previous slice fully consumed
#pragma unroll
        for (int e = 0; e < 8; ++e) {
            const int idx  = e * 256 + tid;          // 2048 float2 entries
            const int step = idx >> 5;               // 0..63
            const int l    = idx & 31;
            const int k0   = j01 * 256 + step * 4 + skap;
            float b0 = scaug, b1 = scaug;
            if (slt7) {
                b0 = coeff[k0 * 7 + sn];
                b1 = coeff[(k0 + 1) * 7 + sn];
            }
            Bfrag[step][l] = make_float2(b0, b1);
        }
        __syncthreads();

        // prefetch next slice's coeff rows into cache (256 rows * 28B)
        if (j01 < 15)
            __builtin_prefetch(coeff + (j01 + 1) * 256 * 7 + tid * 7, 0, 1);

        const float a01 = A01[wave][m][j01];
#pragma unroll
        for (int j23 = 0; j23 < 16; ++j23) {
            const float a23 = A23[wave][m][j23];
            const float t   = a01 * a23;
#pragma unroll
            for (int s = 0; s < 4; ++s) {
                const float2 bf = Bfrag[j23 * 4 + s][lane];   // ds_load_b64
                v2f bv; bv.x = bf.x; bv.y = bf.y;
                v2f av; av.x = t * e0[s]; av.y = t * e1[s];
                acc = __builtin_amdgcn_wmma_f32_16x16x4_f32(
                    false, av, false, bv, (short)0, acc, false, false);
            }
        }
    }

    // ---- epilogue: stage C tile, per-row weighted dot + clipped divide ----
#pragma unroll
    for (int v = 0; v < 8; ++v)
        Sst[wave][v + half * 8][m] = acc[v];

    __syncthreads();

    if (lane < 16) {
        float num = Sst[wave][m][6];                 // constant-term column
#pragma unroll
        for (int i = 0; i < N_IN; ++i)
            num += Sst[wave][m][i] * xv[i];
        float denom = fmaxf(Sst[wave][m][7], 1e-12f);
        out[row] = num / denom;
    }
}

extern "C" void kernel_launch(void* const* d_in, const int* in_sizes, int n_in,
                              void* d_out, int out_size, void* d_ws, size_t ws_size,
                              hipStream_t stream) {
    const float* x     = (const float*)d_in[0];
    const float* mfa   = (const float*)d_in[1];
    const float* mfb   = (const float*)d_in[2];
    const float* mfc   = (const float*)d_in[3];
    const float* coeff = (const float*)d_in[4];
    // d_in[5] = mf_indices: full base-4 enumeration, regenerated arithmetically
    float* out = (float*)d_out;

    dim3 grid(BATCH / (WAVES * TILE_M));   // 32 workgroups
    dim3 block(WAVES * 32);                // 8 wave32 waves
    anfis_wmma_kernel<<<grid, block, 0, stream>>>(x, mfa, mfb, mfc, coeff, out);
}